// MultiLabelSparsemaxLoss_11940009083409
// MI455X (gfx1250) — compile-verified
//
#include <hip/hip_runtime.h>
#include <hip/hip_bf16.h>
#include <math.h>

// ---------------------------------------------------------------------------
// MultiLabelSparsemaxLoss for MI455X (gfx1250, wave32)
//   B=4096 rows, C=32000 classes. HBM-bound: ~1.05 GB => ~45us @23.3TB/s.
//   One workgroup (32 wave32 waves) per row:
//     A) async-DMA the row into LDS (GLOBAL_LOAD_ASYNC_TO_LDS_B128)
//     B) fused max+sum pass
//     C) sort-free Michelot iteration for the sparsemax threshold tau.
//        tau is monotone non-decreasing => active sets are NESTED, so after
//        two full passes we deterministically compact the ~K survivors into a
//        small LDS buffer and iterate over ~7k elements instead of 32k.
//     D) fused loss pass streaming target (non-temporal) and weight (L2).
//   Deterministic: fixed reduction trees, scan-ordered compaction, no atomics.
// ---------------------------------------------------------------------------

typedef int   v4i __attribute__((vector_size(16)));               // 16B payload
typedef float v4f __attribute__((vector_size(16)));               // native vec4
typedef __attribute__((address_space(1))) v4i* gv4i_p;            // global
typedef __attribute__((address_space(3))) v4i* lv4i_p;            // LDS
typedef __attribute__((address_space(1))) void* gvoid_p;
typedef __attribute__((address_space(3))) void* lvoid_p;

#define BLOCK_THREADS 1024   // 32 wave32 waves
#define NWAVES 32
#define COMPACT_CAP 7680     // floats; 30KB. Gaussian rows: |{z>tau1}| ~ 6.8k

__device__ __forceinline__ float wave_red_sum(float v) {
#pragma unroll
  for (int o = 16; o > 0; o >>= 1) v += __shfl_xor(v, o, 32);
  return v;
}

__device__ __forceinline__ float wave_red_max(float v) {
#pragma unroll
  for (int o = 16; o > 0; o >>= 1) v = fmaxf(v, __shfl_xor(v, o, 32));
  return v;
}

// Reduce (a with max-or-sum, b with sum) across a 1024-thread block.
// scr must hold >= 64 floats. Result broadcast to all threads.
__device__ __forceinline__ float2 block_reduce2(float a, float b, float* scr,
                                                int tid, bool a_is_max) {
  const int lane = tid & 31;
  const int wid  = tid >> 5;
  a = a_is_max ? wave_red_max(a) : wave_red_sum(a);
  b = wave_red_sum(b);
  __syncthreads();  // protect scratch against previous use (WAR)
  if (lane == 0) { scr[wid] = a; scr[NWAVES + wid] = b; }
  __syncthreads();
  if (wid == 0) {
    float x = scr[lane];
    float y = scr[NWAVES + lane];
    x = a_is_max ? wave_red_max(x) : wave_red_sum(x);
    y = wave_red_sum(y);
    if (lane == 0) { scr[0] = x; scr[NWAVES] = y; }
  }
  __syncthreads();
  float2 r;
  r.x = scr[0];
  r.y = scr[NWAVES];
  return r;
}

// Deterministic block-wide exclusive scan of per-thread int counts.
// iscr must hold >= 32 ints.
__device__ __forceinline__ int block_excl_scan(int v, int tid, int* iscr) {
  const int lane = tid & 31;
  const int wid  = tid >> 5;
  int x = v;
#pragma unroll
  for (int o = 1; o < 32; o <<= 1) {
    int y = __shfl_up(x, (unsigned)o, 32);
    if (lane >= o) x += y;
  }                                   // x = wave-inclusive scan
  __syncthreads();
  if (lane == 31) iscr[wid] = x;      // wave totals
  __syncthreads();
  if (wid == 0) {
    int w = (lane < NWAVES) ? iscr[lane] : 0;
#pragma unroll
    for (int o = 1; o < 32; o <<= 1) {
      int y = __shfl_up(w, (unsigned)o, 32);
      if (lane >= o) w += y;
    }
    iscr[lane] = w;                   // inclusive scan of wave totals
  }
  __syncthreads();
  const int waveBase = (wid == 0) ? 0 : iscr[wid - 1];
  return waveBase + (x - v);          // exclusive offset of this thread
}

__global__ void __launch_bounds__(BLOCK_THREADS)
sparsemax_row_kernel(const float* __restrict__ inp,
                     const float* __restrict__ tgt,
                     const float* __restrict__ wgt,
                     float* __restrict__ rowloss,
                     int C) {
  extern __shared__ float smem[];
  float* sdata = smem;                      // C floats: raw input row
  float* scr   = smem + C;                  // 64 floats: reduction scratch
  int*   iscr  = (int*)(smem + C + 64);     // 32 ints: scan scratch
  float* cbuf  = smem + C + 64 + 32;        // COMPACT_CAP floats: survivors

  const int tid  = threadIdx.x;
  const int nthr = blockDim.x;
  const int row  = blockIdx.x;
  const int nvec = C >> 2;

  const float*  grow  = inp + (size_t)row * (size_t)C;
  const float4* grow4 = (const float4*)grow;
  float4*       sv    = (float4*)sdata;

  // ---- Phase A: stage row into LDS via gfx1250 async DMA ------------------
#if defined(__gfx1250__) && __has_builtin(__builtin_amdgcn_global_load_async_to_lds_b128)
  for (int j = tid; j < nvec; j += nthr) {
    __builtin_amdgcn_global_load_async_to_lds_b128(
        (gv4i_p)(gvoid_p)(grow4 + j),   // global src: AS(1) v4i*
        (lv4i_p)(lvoid_p)(sv + j),      // LDS dst:    AS(3) v4i*
        0, 0);
  }
#if __has_builtin(__builtin_amdgcn_s_wait_asynccnt)
  __builtin_amdgcn_s_wait_asynccnt(0);
#else
  asm volatile("s_wait_asynccnt 0" ::: "memory");
#endif
#else
  for (int j = tid; j < nvec; j += nthr) sv[j] = grow4[j];
#endif
  for (int i = (nvec << 2) + tid; i < C; i += nthr) sdata[i] = grow[i];  // tail
  __syncthreads();

  // ---- Phase B: row max and raw sum in a single LDS pass ------------------
  float m = -3.402823466e38f;
  float s = 0.0f;
  for (int j = tid; j < nvec; j += nthr) {
    float4 v = sv[j];
    m = fmaxf(fmaxf(m, v.x), fmaxf(v.y, fmaxf(v.z, v.w)));
    s += (v.x + v.y) + (v.z + v.w);
  }
  for (int i = (nvec << 2) + tid; i < C; i += nthr) {
    float v = sdata[i];
    m = fmaxf(m, v);
    s += v;
  }
  float2 ms = block_reduce2(m, s, scr, tid, /*a_is_max=*/true);
  const float rowmax = ms.x;
  // sum(z) = sum(raw) - C*max;  tau0 = (sum(z) - 1)/C  (Michelot init)
  float tau = (ms.y - (float)C * rowmax - 1.0f) / (float)C;

  // ---- Phase C: Michelot fixed point, nested-set compaction ---------------
  // Pass 1 (full set, tau0):
  {
    float ls = 0.0f, lc = 0.0f;
    for (int j = tid; j < nvec; j += nthr) {
      float4 v = sv[j];
      float z0 = v.x - rowmax, z1 = v.y - rowmax;
      float z2 = v.z - rowmax, z3 = v.w - rowmax;
      if (z0 > tau) { ls += z0; lc += 1.0f; }
      if (z1 > tau) { ls += z1; lc += 1.0f; }
      if (z2 > tau) { ls += z2; lc += 1.0f; }
      if (z3 > tau) { ls += z3; lc += 1.0f; }
    }
    for (int i = (nvec << 2) + tid; i < C; i += nthr) {
      float z = sdata[i] - rowmax;
      if (z > tau) { ls += z; lc += 1.0f; }
    }
    float2 sk = block_reduce2(ls, lc, scr, tid, false);
    tau = (sk.x - 1.0f) / sk.y;                       // tau1 (K>=1 always)
  }

  // Pass 2 (full set, tau1) + per-thread survivor count for compaction:
  const float tau1 = tau;
  int   cnt = 0;
  bool  converged = false;
  {
    float ls = 0.0f;
    for (int j = tid; j < nvec; j += nthr) {
      float4 v = sv[j];
      float z0 = v.x - rowmax, z1 = v.y - rowmax;
      float z2 = v.z - rowmax, z3 = v.w - rowmax;
      if (z0 > tau1) { ls += z0; ++cnt; }
      if (z1 > tau1) { ls += z1; ++cnt; }
      if (z2 > tau1) { ls += z2; ++cnt; }
      if (z3 > tau1) { ls += z3; ++cnt; }
    }
    for (int i = (nvec << 2) + tid; i < C; i += nthr) {
      float z = sdata[i] - rowmax;
      if (z > tau1) { ls += z; ++cnt; }
    }
    float2 sk = block_reduce2(ls, (float)cnt, scr, tid, false);
    const float tau2 = (sk.x - 1.0f) / sk.y;
    const int K1 = (int)sk.y;                          // |{z > tau1}|, exact
    converged = (tau2 == tau1);

    const bool useCompact = (K1 <= COMPACT_CAP);       // uniform decision
    int nAct = C;
    if (useCompact && !converged) {
      // Deterministic compaction of {z > tau1} (ordered by owner thread).
      int w = block_excl_scan(cnt, tid, iscr);
      for (int j = tid; j < nvec; j += nthr) {
        float4 v = sv[j];
        float z0 = v.x - rowmax, z1 = v.y - rowmax;
        float z2 = v.z - rowmax, z3 = v.w - rowmax;
        if (z0 > tau1) cbuf[w++] = z0;
        if (z1 > tau1) cbuf[w++] = z1;
        if (z2 > tau1) cbuf[w++] = z2;
        if (z3 > tau1) cbuf[w++] = z3;
      }
      for (int i = (nvec << 2) + tid; i < C; i += nthr) {
        float z = sdata[i] - rowmax;
        if (z > tau1) cbuf[w++] = z;
      }
      nAct = K1;
      __syncthreads();                                 // cbuf visible
    }

    if (!converged) {
      tau = tau2;
      for (int it = 0; it < 62; ++it) {
        float ls2 = 0.0f, lc2 = 0.0f;
        if (useCompact) {
          for (int i = tid; i < nAct; i += nthr) {     // z pre-shifted
            float z = cbuf[i];
            if (z > tau) { ls2 += z; lc2 += 1.0f; }
          }
        } else {
          for (int j = tid; j < nvec; j += nthr) {
            float4 v = sv[j];
            float z0 = v.x - rowmax, z1 = v.y - rowmax;
            float z2 = v.z - rowmax, z3 = v.w - rowmax;
            if (z0 > tau) { ls2 += z0; lc2 += 1.0f; }
            if (z1 > tau) { ls2 += z1; lc2 += 1.0f; }
            if (z2 > tau) { ls2 += z2; lc2 += 1.0f; }
            if (z3 > tau) { ls2 += z3; lc2 += 1.0f; }
          }
          for (int i = (nvec << 2) + tid; i < C; i += nthr) {
            float z = sdata[i] - rowmax;
            if (z > tau) { ls2 += z; lc2 += 1.0f; }
          }
        }
        float2 sk2 = block_reduce2(ls2, lc2, scr, tid, false);
        float ntau = (sk2.x - 1.0f) / sk2.y;
        if (ntau == tau) break;                        // uniform: barrier-safe
        tau = ntau;
      }
    }
  }

  // ---- Phase D: fused loss pass (target streamed NT, weight L2) -----------
  const float tau2q = tau * tau;
  const v4f*    trow = (const v4f*)(tgt + (size_t)row * (size_t)C);
  const float4* wv   = (const float4*)wgt;
  float acc = 0.0f;
  for (int j = tid; j < nvec; j += nthr) {
    float4 v = sv[j];
    v4f    t = __builtin_nontemporal_load(trow + j);   // read-once stream
    float4 w = wv[j];
    float z0 = v.x - rowmax, z1 = v.y - rowmax;
    float z2 = v.z - rowmax, z3 = v.w - rowmax;
    float l0 = 0.5f * (fmaxf(z0 * z0 - tau2q, 0.0f) + t[0] * t[0]) - t[0] * z0;
    float l1 = 0.5f * (fmaxf(z1 * z1 - tau2q, 0.0f) + t[1] * t[1]) - t[1] * z1;
    float l2 = 0.5f * (fmaxf(z2 * z2 - tau2q, 0.0f) + t[2] * t[2]) - t[2] * z2;
    float l3 = 0.5f * (fmaxf(z3 * z3 - tau2q, 0.0f) + t[3] * t[3]) - t[3] * z3;
    acc += (w.x * l0 + w.y * l1) + (w.z * l2 + w.w * l3);
  }
  for (int i = (nvec << 2) + tid; i < C; i += nthr) {
    float z = sdata[i] - rowmax;
    float t = tgt[(size_t)row * (size_t)C + i];
    float l = 0.5f * (fmaxf(z * z - tau2q, 0.0f) + t * t) - t * z;
    acc += wgt[i] * l;
  }
  float2 tot = block_reduce2(acc, 0.0f, scr, tid, false);
  if (tid == 0) rowloss[row] = tot.x;
}

// Deterministic final mean over B per-row losses.
__global__ void __launch_bounds__(BLOCK_THREADS)
mean_kernel(const float* __restrict__ rowloss, float* __restrict__ out, int B) {
  __shared__ float scr[64];
  const int tid = threadIdx.x;
  float s = 0.0f;
  for (int i = tid; i < B; i += blockDim.x) s += rowloss[i];
  float2 tot = block_reduce2(s, 0.0f, scr, tid, false);
  if (tid == 0) out[0] = tot.x / (float)B;
}

extern "C" void kernel_launch(void* const* d_in, const int* in_sizes, int n_in,
                              void* d_out, int out_size, void* d_ws, size_t ws_size,
                              hipStream_t stream) {
  (void)n_in; (void)out_size; (void)ws_size;
  const float* inp = (const float*)d_in[0];
  const float* tgt = (const float*)d_in[1];
  const float* wgt = (const float*)d_in[2];
  const int C = in_sizes[2];          // weight length = #classes
  const int B = in_sizes[0] / C;      // rows
  float* out = (float*)d_out;
  float* rowloss = (float*)d_ws;      // B floats of scratch

  const size_t shmem =
      ((size_t)C + 64 + 32 + COMPACT_CAP) * sizeof(float);  // ~159 KB
  // Allow >64KB dynamic LDS (gfx1250 WGP has 320KB => 2 workgroups/WGP).
  (void)hipFuncSetAttribute((const void*)sparsemax_row_kernel,
                            hipFuncAttributeMaxDynamicSharedMemorySize,
                            (int)shmem);

  sparsemax_row_kernel<<<B, BLOCK_THREADS, shmem, stream>>>(inp, tgt, wgt,
                                                            rowloss, C);
  mean_kernel<<<1, BLOCK_THREADS, 0, stream>>>(rowloss, out, B);
}